// MultiHeadAttentionWithKVCache_51848845197689
// MI455X (gfx1250) — compile-verified
//
#include <hip/hip_runtime.h>
#include <cstdint>

// ---------------------------------------------------------------------------
// MHA with KV cache, MI455X (gfx1250), wave32.
// Memory-bound: must stream 1 GiB of K/V cache once -> ~46us @ 23.3 TB/s.
// fp32 WMMA (v_wmma_f32_16x16x4_f32) for QK^T and P*V.
// K/V tiles streamed to LDS by the Tensor Data Mover (TENSORcnt) when the
// builtin exists, else async-to-LDS (ASYNCcnt); double-buffered either way.
// ---------------------------------------------------------------------------

#define B_      32
#define S_      4
#define E_      1024
#define H_      16
#define D_      64
#define KV_     4096
#define NSPLIT  8
#define CHUNK   (KV_ / NSPLIT)     // 512 kv rows per workgroup
#define NTILES  (CHUNK / 16)       // 32 tiles of 16 rows
#define KSTRIDE 68                 // padded LDS row stride (floats): conflict-free
#define PSTRIDE 18
#define TILEB   (16 * KSTRIDE * 4) // bytes per K/V tile buffer (4352, 16B aligned)

typedef float v2f __attribute__((ext_vector_type(2)));
typedef float v8f __attribute__((ext_vector_type(8)));
typedef unsigned int v4u __attribute__((ext_vector_type(4)));
typedef int v4i __attribute__((ext_vector_type(4)));
typedef int v8i __attribute__((ext_vector_type(8)));

#if __has_builtin(__builtin_amdgcn_tensor_load_to_lds) && \
    __has_builtin(__builtin_amdgcn_s_wait_tensorcnt)
#define USE_TDM 1
#else
#define USE_TDM 0
#endif

__device__ __forceinline__ void wait_ds0() {
  asm volatile("s_wait_dscnt 0x0" ::: "memory");
}

#if USE_TDM
// Tensor DMA: load `rows` x 64 f32 (row stride 256B in memory) into LDS at
// ldsB, padding each 64-dword row by 4 dwords (-> KSTRIDE=68 floats).
// D# packing per CDNA5 ISA ch.8 (group0 128b, group1 256b; groups 2/3 unused;
// this toolchain's builtin takes an extra zero-filled int32x8 before cpol).
__device__ __forceinline__ void tdm_load_rows(const float* g, uint32_t ldsB, int rows) {
  const unsigned long long ga = (unsigned long long)(uintptr_t)g;
  v4u g0;
  g0[0] = 1u;                                            // count=1, user mode
  g0[1] = ldsB;                                          // lds_addr
  g0[2] = (unsigned)ga;                                  // global_addr[31:0]
  g0[3] = (unsigned)((ga >> 32) & 0x01FFFFFFu) | 0x80000000u; // [56:32] | type=2
  v8i g1;
  // mask=0 | data_size=4B(2<<16) | pad_enable(1<<20) | pad_interval=64dw(5<<22)
  // | pad_amount=4dw(3<<25)
  g1[0] = 0x07520000;
  g1[1] = (int)(64u << 16);                              // tensor_dim0 = 64 elems
  g1[2] = (int)((unsigned)rows << 16);                   // tensor_dim1 = rows
  g1[3] = (int)(64u << 16);                              // tile_dim0 = 64
  g1[4] = rows;                                          // tile_dim1 = rows
  g1[5] = D_;                                            // tensor_dim0_stride = 64
  g1[6] = 0;
  g1[7] = 0;
  v4i zz4 = {0, 0, 0, 0};
  v8i zz8 = {0, 0, 0, 0, 0, 0, 0, 0};
  __builtin_amdgcn_tensor_load_to_lds(g0, g1, zz4, zz4, zz8, 0);
}
#endif

// ===========================================================================
// Kernel 1/4: projection GEMM  y[r, col] = dot(X[r,:], W[col,:]) + b[col]
// ===========================================================================
__global__ __launch_bounds__(128)
void proj_kernel(const float* __restrict__ X, const float* __restrict__ W,
                 const float* __restrict__ bias, float* __restrict__ out,
                 int headReorder) {
  __shared__ float xrow[E_];
  const int r = blockIdx.x;
  const int t = threadIdx.x;

  const float4* xs = (const float4*)(X + (size_t)r * E_);
  float4* xd = (float4*)xrow;
  for (int i = t; i < E_ / 4; i += 128) xd[i] = xs[i];
  __syncthreads();

  const int col = blockIdx.y * 128 + t;
  const float4* w4 = (const float4*)(W + (size_t)col * E_);
  float acc = 0.f;
#pragma unroll 8
  for (int i = 0; i < E_ / 4; ++i) {
    float4 a = ((const float4*)xrow)[i];
    float4 b = w4[i];
    acc += a.x * b.x + a.y * b.y + a.z * b.z + a.w * b.w;
  }
  acc += bias[col];

  if (headReorder) {
    const int b = r / S_, s = r % S_;
    const int h = col >> 6, dd = col & 63;
    out[(((size_t)(b * H_ + h)) * S_ + s) * D_ + dd] = acc;
  } else {
    out[(size_t)r * E_ + col] = acc;
  }
}

// ===========================================================================
// Kernel 2/4: flash-attention partials. 1 wave per block, grid = B*H*NSPLIT.
// fp32 WMMA operand layouts (ISA 7.12.2):
//   A 16x4 : lane L<16 holds A[M=L][K=0..1], lane 16+L holds A[M=L][K=2..3]
//   B 4x16 : lane L<16 holds B[K=0..1][N=L], lane 16+L holds B[K=2..3][N=L]
//   C/D    : vgpr i, lanes 0-15 = row i, lanes 16-31 = row 8+i
// ===========================================================================
__global__ __launch_bounds__(32)
void attn_partial_kernel(const float* __restrict__ cacheK,
                         const float* __restrict__ cacheV,
                         const float* __restrict__ Qp,   // [B,H,S,d]
                         const float* __restrict__ Knp,  // [B,H,S,d]
                         const float* __restrict__ Vnp,  // [B,H,S,d]
                         float* __restrict__ pO,         // [B*H, NSPLIT, S, d]
                         float* __restrict__ pM,         // [B*H, NSPLIT, S]
                         float* __restrict__ pL) {       // [B*H, NSPLIT, S]
  __shared__ float Kt[2 * 16 * KSTRIDE];   // double-buffered K tile
  __shared__ float Vt[2 * 16 * KSTRIDE];   // double-buffered V tile
  __shared__ float Pt[4 * PSTRIDE];        // P transpose staging

  const int lane = threadIdx.x;
  const int lo = lane & 15;
  const int hi = lane >> 4;
  const int sp = blockIdx.x % NSPLIT;
  const int bh = blockIdx.x / NSPLIT;

  const float* Kbase = cacheK + (size_t)bh * KV_ * D_;
  const float* Vbase = cacheV + (size_t)bh * KV_ * D_;
  const float* Qb    = Qp  + (size_t)bh * S_ * D_;
  const float* KnB   = Knp + (size_t)bh * S_ * D_;
  const float* VnB   = Vnp + (size_t)bh * S_ * D_;

  const uint32_t ktOff = (uint32_t)(uintptr_t)&Kt[0];
  const uint32_t vtOff = (uint32_t)(uintptr_t)&Vt[0];

#if !USE_TDM
  // per-lane copy pattern, invariant across tiles: 256 b128 chunks / 32 lanes
  uint32_t ldsrel[8], voffs[8];
#pragma unroll
  for (int j = 0; j < 8; ++j) {
    const int chunk = j * 32 + lane;
    const int row = chunk >> 2, seg = chunk & 3;
    ldsrel[j] = (uint32_t)((row * KSTRIDE + seg * 4) * 4);
    voffs[j]  = (uint32_t)(row * 256 + seg * 16);
  }
#endif

  // ---- preload Q as 16 A-operands (rows >= S_ zero-padded), pre-scaled 1/8
  v2f aq[16];
#pragma unroll
  for (int c = 0; c < 16; ++c) {
    v2f v = {0.f, 0.f};
    if (lo < S_) v = *(const v2f*)(Qb + lo * D_ + c * 4 + 2 * hi);
    aq[c] = v * 0.125f;  // 1/sqrt(64)
  }

  v8f acc[4];
#pragma unroll
  for (int t = 0; t < 4; ++t) acc[t] = (v8f)(0.f);
  const float NEG_INF = -__builtin_huge_valf();
  float mrow[4] = {NEG_INF, NEG_INF, NEG_INF, NEG_INF};
  float lrow[4] = {0.f, 0.f, 0.f, 0.f};

  const int kv0 = sp * CHUNK;

  // ---- tile streaming (double-buffered) ----
  auto issue_tile = [&](int tileIdx, int buf) {
    const float* gk = Kbase + (size_t)(kv0 + tileIdx * 16) * D_;
    const float* gv = Vbase + (size_t)(kv0 + tileIdx * 16) * D_;
    const uint32_t bufB = (uint32_t)(buf * TILEB);
#if USE_TDM
    tdm_load_rows(gk, ktOff + bufB, 16);
    tdm_load_rows(gv, vtOff + bufB, 16);
#else
    const unsigned long long ak = (unsigned long long)(uintptr_t)gk;
    const unsigned long long av = (unsigned long long)(uintptr_t)gv;
#pragma unroll
    for (int j = 0; j < 8; ++j) {
      asm volatile("global_load_async_to_lds_b128 %0, %1, %2"
                   :: "v"(ktOff + bufB + ldsrel[j]), "v"(voffs[j]), "s"(ak)
                   : "memory");
      asm volatile("global_load_async_to_lds_b128 %0, %1, %2"
                   :: "v"(vtOff + bufB + ldsrel[j]), "v"(voffs[j]), "s"(av)
                   : "memory");
    }
#endif
  };
  auto wait_tile = [&](bool one_tile_still_inflight) {
#if USE_TDM
    if (one_tile_still_inflight) __builtin_amdgcn_s_wait_tensorcnt((short)2);
    else                         __builtin_amdgcn_s_wait_tensorcnt((short)0);
    asm volatile("" ::: "memory");
#else
    if (one_tile_still_inflight) asm volatile("s_wait_asynccnt 0x10" ::: "memory");
    else                         asm volatile("s_wait_asynccnt 0x0" ::: "memory");
#endif
  };

  // ---- one 16-kv-row tile: scores WMMA chain, online softmax, PV WMMA ----
  auto tile_compute = [&](int buf, int valid) {
    const float* KtB = &Kt[buf * 16 * KSTRIDE];
    const float* VtB = &Vt[buf * 16 * KSTRIDE];
    v8f sc = (v8f)(0.f);
#pragma unroll
    for (int c = 0; c < 16; ++c) {
      v2f bk = *(const v2f*)(&KtB[lo * KSTRIDE + c * 4 + 2 * hi]);
      sc = __builtin_amdgcn_wmma_f32_16x16x4_f32(false, aq[c], false, bk,
                                                 (short)0, sc, false, false);
    }
    float corr[4];
#pragma unroll
    for (int j = 0; j < 4; ++j) {
      float sj = sc[j];
      if (valid < 16) sj = (lo < valid) ? sj : NEG_INF;
      float tm = sj;
#pragma unroll
      for (int off = 8; off > 0; off >>= 1)
        tm = fmaxf(tm, __shfl_xor(tm, off, 16));
      const float mn = fmaxf(mrow[j], tm);
      corr[j] = __expf(mrow[j] - mn);
      const float p = __expf(sj - mn);
      float rs = p;
#pragma unroll
      for (int off = 8; off > 0; off >>= 1) rs += __shfl_xor(rs, off, 16);
      lrow[j] = lrow[j] * corr[j] + rs;
      mrow[j] = mn;
      if (hi == 0) Pt[j * PSTRIDE + lo] = p;  // stage P for transpose
    }
#pragma unroll
    for (int t = 0; t < 4; ++t) {
      v8f a = acc[t];
      a[0] *= corr[0]; a[1] *= corr[1]; a[2] *= corr[2]; a[3] *= corr[3];
      acc[t] = a;
    }
    // P (4x16) x V (16x64) accumulated into 4 d-tiles
#pragma unroll
    for (int c = 0; c < 4; ++c) {
      v2f ap = {0.f, 0.f};
      if (lo < S_) ap = *(const v2f*)(&Pt[lo * PSTRIDE + c * 4 + 2 * hi]);
#pragma unroll
      for (int t = 0; t < 4; ++t) {
        v2f bv;
        bv.x = VtB[(c * 4 + 2 * hi) * KSTRIDE + t * 16 + lo];
        bv.y = VtB[(c * 4 + 2 * hi + 1) * KSTRIDE + t * 16 + lo];
        acc[t] = __builtin_amdgcn_wmma_f32_16x16x4_f32(false, ap, false, bv,
                                                       (short)0, acc[t], false, false);
      }
    }
  };

  // ---- main software-pipelined loop over this split's KV chunk ----
  issue_tile(0, 0);
  for (int t = 0; t < NTILES; ++t) {
    wait_ds0();                      // buffer (t+1)&1 reads (tile t-1) complete
    const bool more = (t + 1) < NTILES;
    if (more) issue_tile(t + 1, (t + 1) & 1);
    wait_tile(more);                 // tile t's loads landed (in-order counters)
    tile_compute(t & 1, 16);
  }

  // ---- tail: the S_ new K/V rows (projected this step), last split only ----
  if (sp == NSPLIT - 1) {
    wait_ds0();
#if USE_TDM
    tdm_load_rows(KnB, ktOff, S_);
    tdm_load_rows(VnB, vtOff, S_);
#else
    if (lane < 16) {  // 4 rows * 64B = 16 b128 chunks
      const int row = lane >> 2, seg = lane & 3;
      const uint32_t rel = (uint32_t)((row * KSTRIDE + seg * 4) * 4);
      const uint32_t vo = (uint32_t)(row * 256 + seg * 16);
      asm volatile("global_load_async_to_lds_b128 %0, %1, %2"
                   :: "v"(ktOff + rel), "v"(vo),
                      "s"((unsigned long long)(uintptr_t)KnB) : "memory");
      asm volatile("global_load_async_to_lds_b128 %0, %1, %2"
                   :: "v"(vtOff + rel), "v"(vo),
                      "s"((unsigned long long)(uintptr_t)VnB) : "memory");
    }
#endif
    wait_tile(false);
    tile_compute(0, S_);  // kv rows >= S_ masked to -inf
  }

  // ---- emit partials ----
  const size_t oBase = ((size_t)bh * NSPLIT + sp) * S_ * D_;
#pragma unroll
  for (int t = 0; t < 4; ++t)
#pragma unroll
    for (int j = 0; j < 4; ++j)
      if (hi == 0) pO[oBase + j * D_ + t * 16 + lo] = acc[t][j];
  if (lane == 0) {
#pragma unroll
    for (int j = 0; j < 4; ++j) {
      pM[((size_t)bh * NSPLIT + sp) * S_ + j] = mrow[j];
      pL[((size_t)bh * NSPLIT + sp) * S_ + j] = lrow[j];
    }
  }
}

// ===========================================================================
// Kernel 3/4: log-sum-exp merge of NSPLIT partials -> merged [B,S,E]
// ===========================================================================
__global__ __launch_bounds__(256)
void attn_merge_kernel(const float* __restrict__ pO, const float* __restrict__ pM,
                       const float* __restrict__ pL, float* __restrict__ merged) {
  const int idx = blockIdx.x * 256 + threadIdx.x;  // B*H*S*D
  if (idx >= B_ * H_ * S_ * D_) return;
  const int dd = idx & 63;
  const int q = (idx >> 6) & (S_ - 1);
  const int bh = idx >> 8;
  const int b = bh / H_, h = bh % H_;

  float M = -__builtin_huge_valf();
  for (int sp = 0; sp < NSPLIT; ++sp)
    M = fmaxf(M, pM[((size_t)bh * NSPLIT + sp) * S_ + q]);
  float L = 0.f, O = 0.f;
  for (int sp = 0; sp < NSPLIT; ++sp) {
    const float w = __expf(pM[((size_t)bh * NSPLIT + sp) * S_ + q] - M);
    L += pL[((size_t)bh * NSPLIT + sp) * S_ + q] * w;
    O += pO[(((size_t)bh * NSPLIT + sp) * S_ + q) * D_ + dd] * w;
  }
  merged[((size_t)(b * S_ + q)) * E_ + h * D_ + dd] = O / L;
}

// ===========================================================================
// Host-side launch
// ===========================================================================
extern "C" void kernel_launch(void* const* d_in, const int* in_sizes, int n_in,
                              void* d_out, int out_size, void* d_ws, size_t ws_size,
                              hipStream_t stream) {
  const float* x       = (const float*)d_in[0];
  const float* cache_k = (const float*)d_in[1];
  const float* cache_v = (const float*)d_in[2];
  const float* Wq      = (const float*)d_in[3];
  const float* bq      = (const float*)d_in[4];
  const float* Wk      = (const float*)d_in[5];
  const float* bk      = (const float*)d_in[6];
  const float* Wv      = (const float*)d_in[7];
  const float* bv      = (const float*)d_in[8];
  const float* Wo      = (const float*)d_in[9];
  const float* bo      = (const float*)d_in[10];
  float* out = (float*)d_out;

  float* ws = (float*)d_ws;
  const size_t QKV = (size_t)B_ * H_ * S_ * D_;      // 131072
  float* Q      = ws;
  float* Kn     = Q + QKV;
  float* Vn     = Kn + QKV;
  float* pO     = Vn + QKV;                          // B*H*NSPLIT*S*D
  float* pM     = pO + (size_t)B_ * H_ * NSPLIT * S_ * D_;
  float* pL     = pM + (size_t)B_ * H_ * NSPLIT * S_;
  float* merged = pL + (size_t)B_ * H_ * NSPLIT * S_;

  const dim3 gProj(B_ * S_, E_ / 128), bProj(128);
  proj_kernel<<<gProj, bProj, 0, stream>>>(x, Wq, bq, Q, 1);
  proj_kernel<<<gProj, bProj, 0, stream>>>(x, Wk, bk, Kn, 1);
  proj_kernel<<<gProj, bProj, 0, stream>>>(x, Wv, bv, Vn, 1);

  attn_partial_kernel<<<B_ * H_ * NSPLIT, 32, 0, stream>>>(
      cache_k, cache_v, Q, Kn, Vn, pO, pM, pL);

  attn_merge_kernel<<<(B_ * H_ * S_ * D_ + 255) / 256, 256, 0, stream>>>(
      pO, pM, pL, merged);

  proj_kernel<<<gProj, bProj, 0, stream>>>(merged, Wo, bo, out, 0);
}